// SConv2d_6768868459192
// MI455X (gfx1250) — compile-verified
//
#include <hip/hip_runtime.h>
#include <cstdint>

typedef unsigned int u32;
typedef unsigned long long u64;
typedef __attribute__((ext_vector_type(4))) u32 v4u;
typedef __attribute__((ext_vector_type(8))) int  v8i;
typedef __attribute__((ext_vector_type(4))) int  v4i;

#define N_    2
#define C_    27
#define OC_   32
#define H_    64
#define W_    64
#define PAD_W 66
#define PAD_H 66
#define TILE_X 32
#define HALO_X 34                 // TILE_X + 2
#define P_TOT  (C_ * 3 * HALO_X)  // 2754 patch floats (11 KB LDS)

// maj3(a,b,c) == closed form of the reference's n=3 bipolar majority sim:
//   sum over >=2-subsets of prod(1±x) / 4 - 1  ==  (a+b+c - a*b*c) / 2
__device__ __forceinline__ float maj3(float a, float b, float c) {
  return 0.5f * ((a + b + c) - a * b * c);
}

// Fully unrolled 5-level ternary majority tree over 243 products.
// wrow: wave-uniform global weight row (SMEM s_load path -> SGPR operands).
// s_p : LDS halo tile [c][3][HALO_X]; lane-consecutive conflict-free reads.
__device__ __forceinline__ float maj_tree(const float* __restrict__ wrow,
                                          const float* __restrict__ s_p,
                                          int lane) {
  float u[3];
#pragma unroll
  for (int cu = 0; cu < 3; ++cu) {
    float z[3];
#pragma unroll
    for (int cz = 0; cz < 3; ++cz) {
      float y[3];
#pragma unroll
      for (int cy = 0; cy < 3; ++cy) {
        const int c = (cu * 3 + cz) * 3 + cy;
        float r[3];
#pragma unroll
        for (int i = 0; i < 3; ++i) {
          const float* pp = &s_p[c * (3 * HALO_X) + i * HALO_X + lane];
          const float* ww = &wrow[c * 9 + i * 3];
          r[i] = maj3(pp[0] * ww[0], pp[1] * ww[1], pp[2] * ww[2]);
        }
        y[cy] = maj3(r[0], r[1], r[2]);
      }
      z[cz] = maj3(y[0], y[1], y[2]);
    }
    u[cu] = maj3(z[0], z[1], z[2]);
  }
  return maj3(u[0], u[1], u[2]);
}

// ---- Tensor Data Mover: build D# (CDNA5 ISA) and issue
// tensor_load_to_lds. data_size=4B; tensor dims set huge so OOB-zeroing never
// fires (all source reads are guaranteed in-bounds via the padded staging buf).
__device__ __forceinline__ void tdm_load_tile(u32 lds_byte_off, const void* gptr,
                                              u32 tile0, u32 tile1, u32 tile2,
                                              u64 stride0, u64 stride1) {
  const u64 ga = (u64)(uintptr_t)gptr;
  const u32 T = 1u << 30;  // "infinite" tensor dims -> no OOB zero-fill
  v4u g0;
  g0.x = 1u;                                            // count=1 valid user D#
  g0.y = lds_byte_off;                                  // lds_addr [63:32]
  g0.z = (u32)ga;                                       // global_addr [95:64]
  g0.w = (u32)((ga >> 32) & 0x1FFFFFFull) | (2u << 30); // addr[56:32] | type=2
  v8i g1;
  g1[0] = (int)(2u << 16);                              // data_size=4B; no pad/iter/mask
  g1[1] = (int)((T & 0xFFFFu) << 16);                   // tensor_dim0 lo16
  g1[2] = (int)(((T >> 16) & 0xFFFFu) | ((T & 0xFFFFu) << 16)); // td0 hi | td1 lo
  g1[3] = (int)(((T >> 16) & 0xFFFFu) | (tile0 << 16)); // td1 hi | tile_dim0
  g1[4] = (int)((tile1 & 0xFFFFu) | ((tile2 & 0xFFFFu) << 16)); // tile_dim1|2
  g1[5] = (int)(u32)(stride0 & 0xFFFFFFFFull);          // dim0_stride lo32
  g1[6] = (int)((u32)((stride0 >> 32) & 0xFFFFull)      // dim0_stride hi16
              | (((u32)(stride1 & 0xFFFFull)) << 16));  // dim1_stride lo16
  g1[7] = (int)(u32)((stride1 >> 16) & 0xFFFFFFFFull);  // dim1_stride hi32
  v4i g2; g2[0] = (int)T; g2[1] = 0; g2[2] = 0; g2[3] = 0; // tensor_dim2; tile_dim3=0
  v4i g3; g3[0] = 0; g3[1] = 0; g3[2] = 0; g3[3] = 0;
  v8i g4; g4[0] = 0; g4[1] = 0; g4[2] = 0; g4[3] = 0;   // clang-23 extra operand
  g4[4] = 0; g4[5] = 0; g4[6] = 0; g4[7] = 0;           // (zero-filled form is probe-verified)
  __builtin_amdgcn_tensor_load_to_lds(g0, g1, g2, g3, g4, 0);
}

// ---- Stage 1: zero-padded staging copy x[2,27,64,64] -> xp[2,27,66,66]
__global__ void pad_kernel(const float* __restrict__ x, float* __restrict__ xp,
                           int total) {
  int idx = blockIdx.x * blockDim.x + threadIdx.x;
  if (idx >= total) return;
  int xc = idx % PAD_W; int t = idx / PAD_W;
  int yc = t % PAD_H;   t /= PAD_H;
  int c  = t % C_;      int n = t / C_;
  float v = 0.0f;
  if (xc >= 1 && xc <= W_ && yc >= 1 && yc <= H_)
    v = x[((n * C_ + c) * H_ + (yc - 1)) * W_ + (xc - 1)];
  xp[idx] = v;
}

// ---- Stage 2 (TDM path): block = 32 ocs (wave id) x 32 pixels (lane).
// Wave 0 TDM-loads the 27ch x 3row x 34col halo tile; weights go through the
// scalar (SMEM) path since their address is wave-uniform.
__global__ __launch_bounds__(1024, 1)
void maj_conv_tdm(const float* __restrict__ xp, const float* __restrict__ w,
                  float* __restrict__ out) {
  __shared__ float s_p[P_TOT];  // [c][3][34]
  const int b    = blockIdx.x;       // 256 blocks: n(1b) | h(6b) | strip(1b)
  const int n    = b >> 7;
  const int h    = (b >> 1) & 63;
  const int xs   = (b & 1) * TILE_X;
  const int tid  = (int)threadIdx.x;
  const int lane = tid & 31;   // pixel in strip
  const int oc   = tid >> 5;   // wave id == output channel

  if (tid < 32) {  // one wave issues the DMA (EXEC ignored for tensor ops)
    // input: 3-D tile X=34 (cols), Y=3 (rows), Z=27 (channels) from padded buf
    const float* src = xp + ((u64)n * C_) * (PAD_H * PAD_W) + (u64)h * PAD_W + xs;
    tdm_load_tile((u32)(uintptr_t)(void*)s_p, src, HALO_X, 3, C_,
                  (u64)PAD_W, (u64)(PAD_H * PAD_W));
    __builtin_amdgcn_s_wait_tensorcnt(0);
  }
  __syncthreads();

  // oc is identical across the wave: make it provably scalar so the weight
  // loads select the SMEM path (s_load -> SGPR multiply operands).
  const int roc = __builtin_amdgcn_readfirstlane(oc);
  const float* wrow = w + (u64)roc * 243;

  const float res = maj_tree(wrow, s_p, lane);
  out[((n * OC_ + roc) * H_ + h) * W_ + (xs + lane)] = res;
}

// ---- Fallback (no staging buffer): same layout, bounds-checked LDS fill.
__global__ __launch_bounds__(1024, 1)
void maj_conv_direct(const float* __restrict__ x, const float* __restrict__ w,
                     float* __restrict__ out) {
  __shared__ float s_p[P_TOT];
  const int b    = blockIdx.x;
  const int n    = b >> 7;
  const int h    = (b >> 1) & 63;
  const int xs   = (b & 1) * TILE_X;
  const int tid  = (int)threadIdx.x;
  const int lane = tid & 31;
  const int oc   = tid >> 5;

  for (int i = tid; i < P_TOT; i += 1024) {
    int xi = i % HALO_X; int t = i / HALO_X;
    int ri = t % 3;      int c = t / 3;
    int gy = h - 1 + ri, gx = xs - 1 + xi;
    float v = 0.0f;
    if (gy >= 0 && gy < H_ && gx >= 0 && gx < W_)
      v = x[((n * C_ + c) * H_ + gy) * W_ + gx];
    s_p[i] = v;
  }
  __syncthreads();

  const int roc = __builtin_amdgcn_readfirstlane(oc);
  const float* wrow = w + (u64)roc * 243;

  const float res = maj_tree(wrow, s_p, lane);
  out[((n * OC_ + roc) * H_ + h) * W_ + (xs + lane)] = res;
}

extern "C" void kernel_launch(void* const* d_in, const int* in_sizes, int n_in,
                              void* d_out, int out_size, void* d_ws, size_t ws_size,
                              hipStream_t stream) {
  const float* x  = (const float*)d_in[0];  // [2,27,64,64]
  const float* w  = (const float*)d_in[1];  // [32,27,3,3] == [oc][243]
  float* out      = (float*)d_out;          // [2,32,64,64]
  const int total_pad = N_ * C_ * PAD_H * PAD_W;        // 235224 floats
  const size_t need   = (size_t)total_pad * sizeof(float);
  const int nblocks   = N_ * H_ * (W_ / TILE_X);        // 256

  if (ws_size >= need) {
    float* xp = (float*)d_ws;
    pad_kernel<<<(total_pad + 255) / 256, 256, 0, stream>>>(x, xp, total_pad);
    maj_conv_tdm<<<nblocks, 1024, 0, stream>>>(xp, w, out);
  } else {
    maj_conv_direct<<<nblocks, 1024, 0, stream>>>(x, w, out);
  }
}